// ABC_2D_Agnostic_86904368267931
// MI455X (gfx1250) — compile-verified
//
#include <hip/hip_runtime.h>

// Problem constants (match reference)
#define BATCH 64
#define CHN   32
#define KKR   9             // kernel_size K
#define KPN   64            // kernel_number_per_pixel
#define PIX   1024          // H*W
#define CHW   32768         // C*H*W
#define CK    288           // C*K  (GEMM reduction dim; 72 wmma k-steps of 4)

#define SB_STRIDE 72        // floats; 288B rows (16B-aligned), disjoint half-wave banks
#define SA_STRIDE 292       // floats; 1168B rows (16B-aligned), conflict-free A reads

typedef __attribute__((ext_vector_type(2))) float v2f;
typedef __attribute__((ext_vector_type(8))) float v8f;

#if __has_builtin(__builtin_amdgcn_global_load_async_to_lds_b128) && \
    __has_builtin(__builtin_amdgcn_s_wait_asynccnt)
#define USE_ASYNC 1
typedef int v4i __attribute__((vector_size(16)));
typedef __attribute__((address_space(1))) v4i* gv4i_p;  // global (AS1) v4i*
typedef __attribute__((address_space(3))) v4i* lv4i_p;  // LDS    (AS3) v4i*
#else
#define USE_ASYNC 0
#endif

// ---------------------------------------------------------------------------
// Kernel 1: transpose x [BATCH][CHW] -> xT [CHW][BATCH] so each gathered row
// is 64 contiguous floats (coalesced 256B reads, fully L2-resident: 8MB).
// ---------------------------------------------------------------------------
__global__ __launch_bounds__(256) void xpose_kernel(const float* __restrict__ x,
                                                    float* __restrict__ xT) {
    __shared__ float tile[64][65];
    const int p0 = blockIdx.x * 64;
    const int t  = threadIdx.x;
#pragma unroll
    for (int i = 0; i < 16; ++i) {          // coalesced along p
        int e = i * 256 + t;
        int b = e >> 6;
        int p = e & 63;
        tile[p][b] = x[b * CHW + p0 + p];
    }
    __syncthreads();
#pragma unroll
    for (int i = 0; i < 16; ++i) {          // coalesced along b
        int e = i * 256 + t;
        int p = e >> 6;
        int b = e & 63;
        xT[(p0 + p) * BATCH + b] = tile[p][b];
    }
}

// ---------------------------------------------------------------------------
// Kernel 2: one workgroup per pixel. 64x64 output tile (M=KP, N=batch),
// K=288 reduced with V_WMMA_F32_16X16X4_F32. Full A (weights) and gathered B
// staged in LDS once (~155KB -> 2 blocks/WGP), single barrier before compute.
// Staging uses GLOBAL_LOAD_ASYNC_TO_LDS_B128 (ASYNCcnt) when available.
// ---------------------------------------------------------------------------
__global__ __launch_bounds__(256) void hash_conv_wmma_kernel(
    const float* __restrict__ xT,        // [CHW][BATCH]
    const int*   __restrict__ hash_idx,  // [CHN][PIX][KKR]
    const float* __restrict__ weights,   // [KPN][CK]
    float*       __restrict__ out)       // [BATCH][KPN][PIX]
{
    __shared__ float sA[KPN * SA_STRIDE]; // weights [64][288] (+pad)
    __shared__ float sB[CK * SB_STRIDE];  // gathered [288][64] (+pad)
    __shared__ int   sIdx[CK];            // per-pixel gather indices

    const int pix  = blockIdx.x;
    const int t    = threadIdx.x;
    const int lane = t & 31;
    const int wave = t >> 5;
    const int half = lane >> 4;           // 0: lanes 0-15, 1: lanes 16-31
    const int l    = lane & 15;

    // Stage the 288 gather indices for this pixel (row r = c*9 + k).
    if (t < CK) {
        int c = t / KKR;
        int k = t - c * KKR;
        sIdx[t] = hash_idx[(c * PIX + pix) * KKR + k];
    }
    __syncthreads();

#if USE_ASYNC
    // Gather B: 288 rows x 64 floats = 4608 16B-chunks, 18 per thread.
#pragma unroll
    for (int i = 0; i < 18; ++i) {
        int e  = i * 256 + t;
        int r  = e >> 4;                  // row (0..287)
        int cg = (e & 15) << 2;           // float column (multiple of 4)
        const float* gp = xT + (size_t)sIdx[r] * BATCH + cg;
        float*       lp = sB + r * SB_STRIDE + cg;
        __builtin_amdgcn_global_load_async_to_lds_b128((gv4i_p)gp, (lv4i_p)lp, 0, 0);
    }
    // Load A: 64 rows x 288 floats = 4608 16B-chunks, 18 per thread.
#pragma unroll
    for (int i = 0; i < 18; ++i) {
        int e = i * 256 + t;
        int m = e / 72;                   // 72 16B-chunks per row
        int j = (e - m * 72) << 2;        // float column (multiple of 4)
        const float* gp = weights + m * CK + j;
        float*       lp = sA + m * SA_STRIDE + j;
        __builtin_amdgcn_global_load_async_to_lds_b128((gv4i_p)gp, (lv4i_p)lp, 0, 0);
    }
    __builtin_amdgcn_s_wait_asynccnt(0);  // this wave's async ops done
#else
    // Fallback: synchronous staging through VGPRs.
#pragma unroll
    for (int i = 0; i < (CK * BATCH) / 256; ++i) {   // 72 iters
        int e = i * 256 + t;
        int r = e >> 6;
        int b = e & 63;
        sB[r * SB_STRIDE + b] = xT[(size_t)sIdx[r] * BATCH + b];
    }
#pragma unroll
    for (int i = 0; i < (KPN * CK) / 256; ++i) {     // 72 iters
        int e = i * 256 + t;
        int m = e / CK;
        int j = e - m * CK;
        sA[m * SA_STRIDE + j] = weights[m * CK + j];
    }
#endif
    __syncthreads();   // all waves waited on their own async ops -> LDS ready

    const int m_base  = (wave & 3) * 16;
    const int n_base0 = (wave >> 2) * 32;
    const int n_base1 = n_base0 + 16;

    v8f acc0 = {};
    v8f acc1 = {};

    // 72 k-steps of V_WMMA_F32_16X16X4_F32; A fragment shared by both N-tiles.
#pragma unroll 8
    for (int ks = 0; ks < CK; ks += 4) {
        const int ka = ks + 2 * half;     // ISA layout: K = 2*half + reg
        v2f a, b0, b1;
        a.x  = sA[(m_base + l) * SA_STRIDE + ka];
        a.y  = sA[(m_base + l) * SA_STRIDE + ka + 1];
        b0.x = sB[(ka    ) * SB_STRIDE + n_base0 + l];
        b0.y = sB[(ka + 1) * SB_STRIDE + n_base0 + l];
        b1.x = sB[(ka    ) * SB_STRIDE + n_base1 + l];
        b1.y = sB[(ka + 1) * SB_STRIDE + n_base1 + l];
        acc0 = __builtin_amdgcn_wmma_f32_16x16x4_f32(
                   false, a, false, b0, (short)0, acc0, false, false);
        acc1 = __builtin_amdgcn_wmma_f32_16x16x4_f32(
                   false, a, false, b1, (short)0, acc1, false, false);
    }

    // Store D tiles: out[b][kp][pix]; 192MB L2 absorbs the partial lines.
#pragma unroll
    for (int r = 0; r < 8; ++r) {
        int m = m_base + r + 8 * half;    // D layout: VGPR r -> M = r + 8*half
        out[((size_t)(n_base0 + l) * KPN + m) * PIX + pix] = acc0[r];
        out[((size_t)(n_base1 + l) * KPN + m) * PIX + pix] = acc1[r];
    }
}

// ---------------------------------------------------------------------------
extern "C" void kernel_launch(void* const* d_in, const int* in_sizes, int n_in,
                              void* d_out, int out_size, void* d_ws, size_t ws_size,
                              hipStream_t stream) {
    (void)in_sizes; (void)n_in; (void)out_size; (void)ws_size;

    const float* x        = (const float*)d_in[0];   // [64, 32, 32, 32] f32
    const int*   hash_idx = (const int*)d_in[1];     // [32, 32, 32, 9] i32
    const float* weights  = (const float*)d_in[2];   // [64, 288] f32
    float*       out      = (float*)d_out;           // [64, 64, 32, 32] f32
    float*       xT       = (float*)d_ws;            // [32768, 64] f32 (8 MB)

    xpose_kernel<<<CHW / 64, 256, 0, stream>>>(x, xT);
    hash_conv_wmma_kernel<<<PIX, 256, 0, stream>>>(xT, hash_idx, weights, out);
}